// MultiSpringBalanceLoss_69870527971928
// MI455X (gfx1250) — compile-verified
//
#include <hip/hip_runtime.h>
#include <hip/hip_bf16.h>
#include <math.h>

// MultiSpringBalanceLoss on MI455X (gfx1250).
// Two fused HBM passes (512MB total ~= 22us at 23.3 TB/s) using the Tensor
// Data Mover to DMA 128x128 f32 tiles into LDS (with TDM row padding for
// bank-conflict-free row+column phases). Deterministic cross-block combining:
// uint-mapped atomic min/max (pass 1) and Q32.32 fixed-point u64 atomicAdd
// (pass 2) -- both exactly order-independent.

#define BATCH 8192
#define TS    128
#define TSP   129           // LDS row stride (TDM pads 1 DWORD per 128 DWORDs)
#define G1F   10.0f
#define G2F   0.5f
#define DELTAF 0.2f

typedef unsigned int       u32x4 __attribute__((ext_vector_type(4)));
typedef int                i32x8 __attribute__((ext_vector_type(8)));
typedef int                i32x4 __attribute__((ext_vector_type(4)));
typedef unsigned long long u64;

// ---- monotonic float<->uint mapping (order-preserving, so atomic min/max
// on the uint image is exact and order-independent) ----
__device__ __forceinline__ unsigned fmap(float x) {
  unsigned b = __float_as_uint(x);
  return (b & 0x80000000u) ? ~b : (b | 0x80000000u);
}
__device__ __forceinline__ float funmap(unsigned u) {
  return __uint_as_float((u & 0x80000000u) ? (u ^ 0x80000000u) : ~u);
}

// ---- TDM: DMA one TS x TS f32 tile (global row stride = BATCH floats) into
// LDS, padding each 128-DWORD row by 1 DWORD -> LDS row stride TSP floats.
// Descriptor layout per CDNA5 ISA ch.8 (D# groups 0..3). This toolchain's
// builtin takes 6 args; the extra int32x8 beyond the architectural D# groups
// is passed as zeros ("reserved: set to 0").
__device__ __forceinline__ void tdm_load_tile(const float* gsrc, float* lds_dst) {
  unsigned lds_off = (unsigned)(unsigned long long)lds_dst; // low 32b of flat LDS addr = LDS offset
  unsigned long long ga = (unsigned long long)gsrc;
  u32x4 g0;
  g0.x = 1u;                                            // count=1, user descriptor
  g0.y = lds_off;                                       // lds_addr
  g0.z = (unsigned)ga;                                  // global_addr[31:0]
  g0.w = (unsigned)((ga >> 32) & 0x1FFFFFFull)          // global_addr[56:32]
       | (2u << 30);                                    // type = 2 ("image")
  i32x8 g1;
  g1[0] = (int)((2u << 16)        // data_size = 4 bytes
              | (1u << 20)        // pad_enable
              | (6u << 22)        // pad_interval: every 128 DWORDs
              | (0u << 25));      // pad_amount: 1 DWORD
  g1[1] = (int)(((unsigned)BATCH & 0xFFFFu) << 16);                               // tensor_dim0 lo16
  g1[2] = (int)(((unsigned)BATCH >> 16) | (((unsigned)BATCH & 0xFFFFu) << 16));   // dim0 hi16 | dim1 lo16
  g1[3] = (int)(((unsigned)BATCH >> 16) | ((unsigned)TS << 16));                  // dim1 hi16 | tile_dim0
  g1[4] = (int)TS;                                                                // tile_dim1 (tile_dim2=0)
  g1[5] = (int)BATCH;                                                             // tensor_dim0_stride lo32
  g1[6] = 0;                                                                      // stride0 hi16 | stride1 lo16
  g1[7] = (int)(((unsigned long long)BATCH * (unsigned long long)BATCH) >> 16);   // stride1[47:16]
  i32x4 g2 = { 1, 0, 0, 0 };  // tensor_dim2=1 (unused: 2-D tile)
  i32x4 g3 = { 0, 0, 0, 0 };
  i32x8 g4 = { 0, 0, 0, 0, 0, 0, 0, 0 };  // beyond architectural D#: reserved -> 0
  __builtin_amdgcn_tensor_load_to_lds(g0, g1, g2, g3, g4, 0);
}

// ---------------- kernel 0: init workspace ----------------
__global__ void init_kernel(unsigned* rmax, unsigned* rmin, unsigned* cmax, unsigned* cmin,
                            u64* rp, u64* rn, u64* cp, u64* cn) {
  int i = blockIdx.x * blockDim.x + threadIdx.x;
  if (i < BATCH) {
    rmax[i] = fmap(-INFINITY); rmin[i] = fmap(INFINITY);
    cmax[i] = fmap(-INFINITY); cmin[i] = fmap(INFINITY);
    rp[i] = 0ull; rn[i] = 0ull; cp[i] = 0ull; cn[i] = 0ull;
  }
}

// ---------------- kernel 1: fused row+col hardest-neg / weakest-pos ----------------
__global__ __launch_bounds__(TS) void stats_kernel(
    const float* __restrict__ S, const int* __restrict__ T,
    unsigned* __restrict__ rmax, unsigned* __restrict__ rmin,
    unsigned* __restrict__ cmax, unsigned* __restrict__ cmin) {
  __shared__ float tile[TS * TSP];
  __shared__ int lrow[TS], lcol[TS];
  const int ntx = BATCH / TS;
  int ty = blockIdx.x / ntx, tx = blockIdx.x % ntx;
  int r0 = ty * TS, c0 = tx * TS;
  int t = (int)threadIdx.x;

  if (t < 32) tdm_load_tile(S + (size_t)r0 * BATCH + c0, tile);  // wave0 issues the DMA
  lrow[t] = T[r0 + t];
  lcol[t] = T[c0 + t];
  __builtin_amdgcn_s_wait_tensorcnt(0);
  __syncthreads();

  // row-direction stats for global row r0+t (stride-TSP LDS reads: conflict-free)
  {
    int lr = lrow[t];
    float mx = -INFINITY, mn = INFINITY;
#pragma unroll 8
    for (int k = 0; k < TS; ++k) {
      float s = tile[t * TSP + k];
      if (lr == lcol[k]) mn = fminf(mn, s); else mx = fmaxf(mx, s);
    }
    atomicMax(&rmax[r0 + t], fmap(mx));
    atomicMin(&rmin[r0 + t], fmap(mn));
  }
  // column-direction stats for global column c0+t (consecutive LDS reads)
  {
    int lc = lcol[t];
    float mx = -INFINITY, mn = INFINITY;
#pragma unroll 8
    for (int r = 0; r < TS; ++r) {
      float s = tile[r * TSP + t];
      if (lrow[r] == lc) mn = fminf(mn, s); else mx = fmaxf(mx, s);
    }
    atomicMax(&cmax[c0 + t], fmap(mx));
    atomicMin(&cmin[c0 + t], fmap(mn));
  }
}

__device__ __forceinline__ u64 to_fixed(float v) {
  return (u64)((double)v * 4294967296.0);  // Q32.32
}

// ---------------- kernel 2: fused row+col masked exp-sums ----------------
__global__ __launch_bounds__(TS) void sums_kernel(
    const float* __restrict__ S, const int* __restrict__ T,
    const unsigned* __restrict__ rmax, const unsigned* __restrict__ rmin,
    const unsigned* __restrict__ cmax, const unsigned* __restrict__ cmin,
    u64* __restrict__ rp, u64* __restrict__ rn,
    u64* __restrict__ cp, u64* __restrict__ cn) {
  __shared__ float tile[TS * TSP];
  __shared__ int lrow[TS], lcol[TS];
  const int ntx = BATCH / TS;
  int ty = blockIdx.x / ntx, tx = blockIdx.x % ntx;
  int r0 = ty * TS, c0 = tx * TS;
  int t = (int)threadIdx.x;

  if (t < 32) tdm_load_tile(S + (size_t)r0 * BATCH + c0, tile);
  lrow[t] = T[r0 + t];
  lcol[t] = T[c0 + t];
  // thresholds (independent of the tile DMA)
  float r_thA = funmap(rmax[r0 + t]) + DELTAF;  // ap: S <  max_an + delta
  float r_thB = funmap(rmin[r0 + t]) - DELTAF;  // an: S >  min_ap - delta
  float c_thA = funmap(cmax[c0 + t]) + DELTAF;
  float c_thB = funmap(cmin[c0 + t]) - DELTAF;
  __builtin_amdgcn_s_wait_tensorcnt(0);
  __syncthreads();

  // row phase
  {
    int lr = lrow[t];
    float pos = 0.0f, neg = 0.0f;
#pragma unroll 8
    for (int k = 0; k < TS; ++k) {
      float s = tile[t * TSP + k];
      bool same = (lr == lcol[k]);
      if (same  && s < r_thA) pos += __expf(G2F - G1F * s);
      if (!same && s > r_thB) neg += __expf(G1F * s - G2F);
    }
    u64 p = to_fixed(pos), n = to_fixed(neg);
    if (p) atomicAdd(&rp[r0 + t], p);
    if (n) atomicAdd(&rn[r0 + t], n);
  }
  // column phase
  {
    int lc = lcol[t];
    float pos = 0.0f, neg = 0.0f;
#pragma unroll 8
    for (int r = 0; r < TS; ++r) {
      float s = tile[r * TSP + t];
      bool same = (lrow[r] == lc);
      if (same  && s < c_thA) pos += __expf(G2F - G1F * s);
      if (!same && s > c_thB) neg += __expf(G1F * s - G2F);
    }
    u64 p = to_fixed(pos), n = to_fixed(neg);
    if (p) atomicAdd(&cp[c0 + t], p);
    if (n) atomicAdd(&cn[c0 + t], n);
  }
}

// ---------------- kernel 3: log/valid/mean ----------------
__global__ void final_kernel(const u64* __restrict__ rp, const u64* __restrict__ rn,
                             const u64* __restrict__ cp, const u64* __restrict__ cn,
                             float* __restrict__ out) {
  __shared__ double acc[256];
  double local = 0.0;
  const double inv = 1.0 / 4294967296.0;
  for (int i = (int)threadIdx.x; i < BATCH; i += 256) {
    u64 a = rp[i], b = rn[i], c = cp[i], d = cn[i];
    if (a && b && c && d) {  // valid: all four pair-sets non-empty
      local += log((double)a * inv) + log((double)b * inv)
             + log((double)c * inv) + log((double)d * inv);
    }
  }
  acc[threadIdx.x] = local;
  __syncthreads();
  for (int s = 128; s > 0; s >>= 1) {
    if ((int)threadIdx.x < s) acc[threadIdx.x] += acc[threadIdx.x + s];
    __syncthreads();
  }
  if (threadIdx.x == 0) out[0] = (float)(acc[0] / (10.0 * (double)BATCH)); // /GAMMA_1 /batch
}

extern "C" void kernel_launch(void* const* d_in, const int* in_sizes, int n_in,
                              void* d_out, int out_size, void* d_ws, size_t ws_size,
                              hipStream_t stream) {
  (void)in_sizes; (void)n_in; (void)out_size; (void)ws_size;
  const float* S = (const float*)d_in[0];
  const int*   T = (const int*)d_in[1];
  float* out = (float*)d_out;

  char* ws = (char*)d_ws;
  unsigned* rmax = (unsigned*)(ws + 0 * 32768);
  unsigned* rmin = (unsigned*)(ws + 1 * 32768);
  unsigned* cmax = (unsigned*)(ws + 2 * 32768);
  unsigned* cmin = (unsigned*)(ws + 3 * 32768);
  u64* rp = (u64*)(ws + 131072 + 0 * 65536);
  u64* rn = (u64*)(ws + 131072 + 1 * 65536);
  u64* cp = (u64*)(ws + 131072 + 2 * 65536);
  u64* cn = (u64*)(ws + 131072 + 3 * 65536);

  const int ntiles = (BATCH / TS) * (BATCH / TS);  // 4096

  init_kernel<<<BATCH / 256, 256, 0, stream>>>(rmax, rmin, cmax, cmin, rp, rn, cp, cn);
  stats_kernel<<<ntiles, TS, 0, stream>>>(S, T, rmax, rmin, cmax, cmin);
  sums_kernel<<<ntiles, TS, 0, stream>>>(S, T, rmax, rmin, cmax, cmin, rp, rn, cp, cn);
  final_kernel<<<1, 256, 0, stream>>>(rp, rn, cp, cn, out);
}